// GNN_37598143709464
// MI455X (gfx1250) — compile-verified
//
#include <hip/hip_runtime.h>

typedef __attribute__((ext_vector_type(2))) float v2f;
typedef __attribute__((ext_vector_type(8))) float v8f;

#define F_IN 128
#define HDIM 16

// ---------------- init: zero agg1, deg=1 (self loop), out=0 ----------------
__global__ void gcn_init(float* __restrict__ agg1, float* __restrict__ deg,
                         float* __restrict__ out, int n) {
  int idx = blockIdx.x * blockDim.x + threadIdx.x;
  if (idx < n * HDIM) agg1[idx] = 0.0f;
  if (idx < n) { deg[idx] = 1.0f; out[idx] = 0.0f; }
}

// ---------------- degree: atomic count of dst (self loops pre-seeded) ------
__global__ void gcn_degree(const int* __restrict__ dst, float* __restrict__ deg, int e) {
  int idx = blockIdx.x * blockDim.x + threadIdx.x;
  if (idx < e) atomicAdd(&deg[dst[idx]], 1.0f);
}

// ---------------- dis = rsqrt(max(deg,1)) ---------------------------------
__global__ void gcn_dis(const float* __restrict__ deg, float* __restrict__ dis, int n) {
  int idx = blockIdx.x * blockDim.x + threadIdx.x;
  if (idx < n) dis[idx] = rsqrtf(fmaxf(deg[idx], 1.0f));
}

// ---------------- h = x @ W1 via V_WMMA_F32_16X16X4_F32 -------------------
// One wave32 per 16-row tile. A: 16x4 f32 (2 VGPR), B: 4x16 f32 (2 VGPR),
// C/D: 16x16 f32 (8 VGPR). Layouts per CDNA5 ISA 7.12.2.
__global__ void gcn_gemm1_wmma(const float* __restrict__ x,
                               const float* __restrict__ W1,
                               float* __restrict__ h, int numTiles) {
  int wave = (blockIdx.x * blockDim.x + threadIdx.x) >> 5;
  int lane = threadIdx.x & 31;
  if (wave >= numTiles) return;  // wave-uniform guard: EXEC stays all-1s

  const int row0 = wave * 16;
  const int m    = lane & 15;        // A row / B col for this lane
  const int hi   = lane >> 4;        // 0: K={0,1}, 1: K={2,3}
  const int koff = hi << 1;

  const float* xrow = x + (size_t)(row0 + m) * F_IN + koff;

  v8f c = {};
#pragma unroll 4
  for (int k = 0; k < F_IN; k += 4) {
    // A fragment: lanes 0-15 hold x[row, k..k+1], lanes 16-31 x[row, k+2..k+3]
    v2f a = *(const v2f*)(xrow + k);
    // B fragment: lane holds W1[k+koff.., n=m]
    v2f b;
    b.x = W1[(k + koff) * HDIM + m];
    b.y = W1[(k + koff + 1) * HDIM + m];
    c = __builtin_amdgcn_wmma_f32_16x16x4_f32(
        /*neg_a=*/false, a, /*neg_b=*/false, b,
        /*c_mod=*/(short)0, c, /*reuse_a=*/false, /*reuse_b=*/false);
  }

  // D layout: VGPR v -> row = v + 8*hi, col = m
#pragma unroll
  for (int v = 0; v < 8; v++) {
    int row = row0 + v + (hi << 3);
    h[(size_t)row * HDIM + m] = c[v];
  }
}

// ---------------- layer-1 edge aggregation: agg1[dst] += h[src]*norm ------
__global__ void gcn_edge_agg1(const int* __restrict__ src, const int* __restrict__ dst,
                              const float* __restrict__ dis, const float* __restrict__ h,
                              float* __restrict__ agg1, int e) {
  int idx = blockIdx.x * blockDim.x + threadIdx.x;
  if (idx >= e) return;
  int s = src[idx];
  int d = dst[idx];
  float norm = dis[s] * dis[d];
  const float4* hs = (const float4*)(h + (size_t)s * HDIM);
  float* ad = agg1 + (size_t)d * HDIM;
#pragma unroll
  for (int q = 0; q < 4; q++) {
    float4 v = hs[q];  // 64B gather, L2-resident
    atomicAdd(&ad[q * 4 + 0], v.x * norm);
    atomicAdd(&ad[q * 4 + 1], v.y * norm);
    atomicAdd(&ad[q * 4 + 2], v.z * norm);
    atomicAdd(&ad[q * 4 + 3], v.w * norm);
  }
}

// -------- finalize1: add self loop + bias, ReLU, fold in W2 -> z ----------
__global__ void gcn_finalize1(const float* __restrict__ agg1, const float* __restrict__ h,
                              const float* __restrict__ dis, const float* __restrict__ b1,
                              const float* __restrict__ W2, float* __restrict__ z, int n) {
  int i = blockIdx.x * blockDim.x + threadIdx.x;
  if (i >= n) return;
  float selfn = dis[i] * dis[i];
  const float* ai = agg1 + (size_t)i * HDIM;
  const float* hi = h + (size_t)i * HDIM;
  float acc = 0.0f;
#pragma unroll
  for (int c = 0; c < HDIM; c++) {
    float v = ai[c] + hi[c] * selfn + b1[c];
    v = fmaxf(v, 0.0f);          // ReLU
    acc = fmaf(v, W2[c], acc);   // layer-2 matmul folded in (H x 1)
  }
  z[i] = acc;
}

// ---------------- layer-2 edge aggregation (scalar) -----------------------
__global__ void gcn_edge_agg2(const int* __restrict__ src, const int* __restrict__ dst,
                              const float* __restrict__ dis, const float* __restrict__ z,
                              float* __restrict__ out, int e) {
  int idx = blockIdx.x * blockDim.x + threadIdx.x;
  if (idx >= e) return;
  int s = src[idx];
  int d = dst[idx];
  atomicAdd(&out[d], z[s] * dis[s] * dis[d]);
}

// ---------------- finalize2: self loop + b2 -------------------------------
__global__ void gcn_finalize2(const float* __restrict__ z, const float* __restrict__ dis,
                              const float* __restrict__ b2, float* __restrict__ out, int n) {
  int i = blockIdx.x * blockDim.x + threadIdx.x;
  if (i >= n) return;
  out[i] += z[i] * dis[i] * dis[i] + b2[0];
}

extern "C" void kernel_launch(void* const* d_in, const int* in_sizes, int n_in,
                              void* d_out, int out_size, void* d_ws, size_t ws_size,
                              hipStream_t stream) {
  const float* x   = (const float*)d_in[0];
  const int*   ei  = (const int*)d_in[1];
  const float* W1  = (const float*)d_in[2];
  const float* b1  = (const float*)d_in[3];
  const float* W2  = (const float*)d_in[4];
  const float* b2  = (const float*)d_in[5];
  float*       out = (float*)d_out;

  const int n = in_sizes[0] / F_IN;     // 100000
  const int e = in_sizes[1] / 2;        // 3200000
  const int* src = ei;
  const int* dst = ei + e;

  // workspace layout (floats): deg[n], dis[n], h[n*16], agg1[n*16], z[n]
  float* ws   = (float*)d_ws;
  float* deg  = ws;
  float* dis  = ws + (size_t)n;
  float* h    = ws + (size_t)2 * n;
  float* agg1 = ws + (size_t)2 * n + (size_t)n * HDIM;
  float* z    = ws + (size_t)2 * n + (size_t)2 * n * HDIM;

  const int B = 256;
  const int numTiles = (n + 15) / 16;   // 6250, exact

  gcn_init<<<(n * HDIM + B - 1) / B, B, 0, stream>>>(agg1, deg, out, n);
  gcn_degree<<<(e + B - 1) / B, B, 0, stream>>>(dst, deg, e);
  gcn_dis<<<(n + B - 1) / B, B, 0, stream>>>(deg, dis, n);

  // 8 waves per 256-thread block, one 16x16 output tile per wave
  const int wavesPerBlock = B / 32;
  gcn_gemm1_wmma<<<(numTiles + wavesPerBlock - 1) / wavesPerBlock, B, 0, stream>>>(
      x, W1, h, numTiles);

  gcn_edge_agg1<<<(e + B - 1) / B, B, 0, stream>>>(src, dst, dis, h, agg1, e);
  gcn_finalize1<<<(n + B - 1) / B, B, 0, stream>>>(agg1, h, dis, b1, W2, z, n);
  gcn_edge_agg2<<<(e + B - 1) / B, B, 0, stream>>>(src, dst, dis, z, out, e);
  gcn_finalize2<<<(n + B - 1) / B, B, 0, stream>>>(z, dis, b2, out, n);
}